// Encoder_1614907703321
// MI455X (gfx1250) — compile-verified
//
#include <hip/hip_runtime.h>

// ---------------------------------------------------------------------------
// 2-layer GCN:  out = ( (A_norm H0 W0 + b0)_relu  -> A_norm (.) W1 + b1 )
// A_norm = D^{-1/2} A D^{-1/2}, norm computed from in-degree on dst.
//
// MI455X reasoning: scatter/gather (~1.2 GB) is the bound (~55us @ 23.3 TB/s);
// agg buffer (25.6 MB) is L2-resident so f32 atomics are cheap. The GEMMs
// (3.3 GFLOP) use fp32 WMMA (v_wmma_f32_16x16x4_f32) to keep full precision
// while running on the matrix pipe. W staged in LDS (64 KB of the 320 KB WGP).
// ---------------------------------------------------------------------------

#define GCN_D 128

typedef __attribute__((ext_vector_type(2))) float v2f;
typedef __attribute__((ext_vector_type(8))) float v8f;

// ---------------- zero fill (float4 vectorized) ----------------
__global__ void k_fill_zero4(float4* __restrict__ p, int n4) {
    int i = blockIdx.x * blockDim.x + threadIdx.x;
    if (i < n4) p[i] = make_float4(0.f, 0.f, 0.f, 0.f);
}

// ---------------- in-degree ----------------
__global__ void k_degree(const int* __restrict__ dst, float* __restrict__ deg,
                         int n_edges) {
    int e = blockIdx.x * blockDim.x + threadIdx.x;
    if (e < n_edges) atomicAdd(&deg[dst[e]], 1.0f);
}

// ---------------- norm = rsqrt(max(deg,1)) ----------------
__global__ void k_norm(const float* __restrict__ deg, float* __restrict__ norm,
                       int n_nodes) {
    int i = blockIdx.x * blockDim.x + threadIdx.x;
    if (i < n_nodes) norm[i] = rsqrtf(fmaxf(deg[i], 1.0f));
}

// ---------------- scatter: agg[dst] += h[src] * norm[src] ----------------
// One edge per wave (32 lanes); each lane moves one float4 chunk of the
// 128-float row (32 lanes * 16 B = 512 B contiguous -> fully coalesced).
__global__ void k_scatter(const float* __restrict__ h,
                          const float* __restrict__ norm,
                          const int* __restrict__ src,
                          const int* __restrict__ dst,
                          float* __restrict__ agg, int n_edges) {
    long long t = (long long)blockIdx.x * blockDim.x + threadIdx.x;
    long long total = (long long)n_edges * 32;
    if (t >= total) return;
    int e    = (int)(t >> 5);
    int lane = (int)(t & 31);
    int s = src[e];
    int d = dst[e];
    float ns = norm[s];
    const float4 v = ((const float4*)(h + (size_t)s * GCN_D))[lane];
    float* ap = agg + (size_t)d * GCN_D + lane * 4;
    atomicAdd(ap + 0, v.x * ns);
    atomicAdd(ap + 1, v.y * ns);
    atomicAdd(ap + 2, v.z * ns);
    atomicAdd(ap + 3, v.w * ns);
}

// ---------------- WMMA GEMM: out = relu?( (agg*norm[row]) @ W + b ) --------
// Block = 256 threads = 8 waves. Each block handles 16 rows; wave w handles
// the 16-column tile [16w,16w+16). W (128x128 f32 = 64 KB) staged in LDS.
// K=128 -> 32 chained v_wmma_f32_16x16x4_f32 per tile.
__global__ __launch_bounds__(256) void k_gemm_wmma(
    const float* __restrict__ A,      // [n_rows,128] aggregated features
    const float* __restrict__ norm,   // [n_rows] dst-side norm
    const float* __restrict__ W,      // [128,128] row-major
    const float* __restrict__ bias,   // [128]
    float* __restrict__ out,          // [n_rows,128]
    int n_rows, int do_relu) {
    __shared__ float Wlds[GCN_D * GCN_D];  // 64 KB of the 320 KB WGP LDS

    // cooperative W load, float4 vectorized
    {
        const float4* Wg = (const float4*)W;
        float4* Wl = (float4*)Wlds;
        for (int i = threadIdx.x; i < GCN_D * GCN_D / 4; i += 256) Wl[i] = Wg[i];
    }
    __syncthreads();

    const int wave = threadIdx.x >> 5;   // 0..7 -> column tile
    const int lane = threadIdx.x & 31;
    const int m    = lane & 15;          // A row within tile (both halves)
    const int half = lane >> 4;          // 0 -> K={0,1}, 1 -> K={2,3}
    const int n    = lane & 15;          // B/C column within tile
    const int row0 = blockIdx.x * 16;
    const int col0 = wave * 16;

    int arow = row0 + m;
    if (arow >= n_rows) arow = n_rows - 1;   // clamp (n_rows=50000 is 16-divisible)
    const float  nrm  = norm[arow];
    const float* Arow = A + (size_t)arow * GCN_D;

    v8f acc = {};
    #pragma unroll 4
    for (int k0 = 0; k0 < GCN_D; k0 += 4) {
        const int ka = k0 + 2 * half;
        v2f a, b;
        a.x = Arow[ka] * nrm;                 // A(m, ka)   * dst-norm
        a.y = Arow[ka + 1] * nrm;             // A(m, ka+1) * dst-norm
        b.x = Wlds[ka * GCN_D + col0 + n];        // W(ka,   col0+n)
        b.y = Wlds[(ka + 1) * GCN_D + col0 + n];  // W(ka+1, col0+n)
        // D = A x B + C, full fp32 (16x16x4)
        acc = __builtin_amdgcn_wmma_f32_16x16x4_f32(
            /*neg_a=*/false, a, /*neg_b=*/false, b,
            /*c_mod=*/(short)0, acc, /*reuse_a=*/false, /*reuse_b=*/false);
    }

    // epilogue: C/D VGPR r holds (M = r + 8*half, N = lane&15)
    const float bv = bias[col0 + n];
    #pragma unroll
    for (int r = 0; r < 8; ++r) {
        const int mr = r + 8 * half;
        const int orow = row0 + mr;
        if (orow < n_rows) {
            float v = acc[r] + bv;
            if (do_relu) v = fmaxf(v, 0.0f);
            out[(size_t)orow * GCN_D + col0 + n] = v;
        }
    }
}

// ---------------------------------------------------------------------------
extern "C" void kernel_launch(void* const* d_in, const int* in_sizes, int n_in,
                              void* d_out, int out_size, void* d_ws, size_t ws_size,
                              hipStream_t stream) {
    const float* features = (const float*)d_in[0];
    const int*   src      = (const int*)d_in[1];
    const int*   dst      = (const int*)d_in[2];
    const float* W0       = (const float*)d_in[3];
    const float* b0       = (const float*)d_in[4];
    const float* W1       = (const float*)d_in[5];
    const float* b1       = (const float*)d_in[6];

    const int n_nodes = in_sizes[0] / GCN_D;
    const int n_edges = in_sizes[1];

    float* out = (float*)d_out;

    // workspace layout: [agg: n_nodes*128 f32][deg: n_nodes f32][norm: n_nodes f32]
    float* agg  = (float*)d_ws;
    float* deg  = agg + (size_t)n_nodes * GCN_D;
    float* norm = deg + n_nodes;

    const int BT = 256;
    const int agg_n4  = n_nodes * GCN_D / 4;
    const int node_n4 = (n_nodes + 3) / 4;   // n_nodes divisible by 4 here
    const long long scat_threads = (long long)n_edges * 32;
    const int scat_blocks = (int)((scat_threads + BT - 1) / BT);
    const int row_tiles = (n_nodes + 15) / 16;

    // ---- degree + norm ----
    k_fill_zero4<<<(node_n4 + BT - 1) / BT, BT, 0, stream>>>((float4*)deg, node_n4);
    k_degree<<<(n_edges + BT - 1) / BT, BT, 0, stream>>>(dst, deg, n_edges);
    k_norm<<<(n_nodes + BT - 1) / BT, BT, 0, stream>>>(deg, norm, n_nodes);

    // ---- layer 0: agg = scatter(features * norm[src]); h1 = relu(agg*norm @ W0 + b0)
    k_fill_zero4<<<(agg_n4 + BT - 1) / BT, BT, 0, stream>>>((float4*)agg, agg_n4);
    k_scatter<<<scat_blocks, BT, 0, stream>>>(features, norm, src, dst, agg, n_edges);
    // stage h1 in d_out (exactly n_nodes*128 floats); fully overwritten below
    k_gemm_wmma<<<row_tiles, BT, 0, stream>>>(agg, norm, W0, b0, out, n_nodes, 1);

    // ---- layer 1: agg = scatter(h1 * norm[src]); out = agg*norm @ W1 + b1
    k_fill_zero4<<<(agg_n4 + BT - 1) / BT, BT, 0, stream>>>((float4*)agg, agg_n4);
    k_scatter<<<scat_blocks, BT, 0, stream>>>(out, norm, src, dst, agg, n_edges);
    k_gemm_wmma<<<row_tiles, BT, 0, stream>>>(agg, norm, W1, b1, out, n_nodes, 0);
}